// GAT_58394375356708
// MI455X (gfx1250) — compile-verified
//
#include <hip/hip_runtime.h>

// GAT 2-layer forward for MI455X (gfx1250, wave32).
// GEMMs: v_wmma_f32_16x16x4_f32, weights staged in LDS via
// global_load_async_to_lds_b128 (ASYNCcnt). Edge softmax/agg: atomics,
// all gather structures L2-resident (192MB L2).

typedef __attribute__((ext_vector_type(2))) float v2f;
typedef __attribute__((ext_vector_type(8))) float v8f;

#define NEG_SLOPE 0.2f

// Monotone (race-safe) float atomic max: int-compare for >=0, uint-compare for <0.
__device__ __forceinline__ void atomic_max_f32(float* addr, float v) {
  if (v >= 0.0f) atomicMax((int*)addr, __float_as_int(v));
  else           atomicMin((unsigned int*)addr, __float_as_uint(v));
}

// Async copy of `bytes` (multiple of 16) from global to LDS, whole block helps.
// lds_off = raw LDS byte offset (low 32 bits of generic shared-space pointer).
__device__ __forceinline__ void async_copy_to_lds(unsigned lds_off,
                                                  const void* gsrc,
                                                  unsigned bytes, int tid, int nthreads) {
  unsigned long long ga = (unsigned long long)(uintptr_t)gsrc;
  for (unsigned off = tid * 16u; off < bytes; off += nthreads * 16u) {
    unsigned l = lds_off + off;
    unsigned long long g = ga + off;
    asm volatile("global_load_async_to_lds_b128 %0, %1, off"
                 :: "v"(l), "v"(g) : "memory");
  }
  asm volatile("s_wait_asynccnt 0x0" ::: "memory");
}

// ---------------- init: m=-inf, denom=0, agg buffers=0 ----------------
__global__ void gat_init(float* __restrict__ m1, float* __restrict__ m2,
                         float* __restrict__ dn1, float* __restrict__ dn2,
                         float* __restrict__ agg1, float* __restrict__ out, int N) {
  int i = blockIdx.x * blockDim.x + threadIdx.x;
  if (i < N) {
    float ninf = __int_as_float(0xFF800000);
    m1[i] = ninf; m2[i] = ninf; dn1[i] = 0.f; dn2[i] = 0.f;
  }
  if (i < 16 * N) agg1[i] = 0.f;
  if (i < 40 * N) out[i]  = 0.f;
}

// ---------------- GEMM1: h1[N,16] = x[N,512] @ W1[512,16] via WMMA ----------------
// One wave per 16-node tile; 128 k-steps of V_WMMA_F32_16X16X4_F32.
// W1 (32KB) staged once per block into LDS with async-to-LDS copies.
__global__ void gemm1_wmma(const float* __restrict__ x, const float* __restrict__ W1,
                           float* __restrict__ h1, int N) {
  __shared__ float sW[512 * 16];
  {
    unsigned lds_off = (unsigned)(uintptr_t)&sW[0];   // truncate -> LDS byte offset
    async_copy_to_lds(lds_off, W1, 512 * 16 * 4, threadIdx.x, blockDim.x);
  }
  __syncthreads();

  int gwave = (blockIdx.x * blockDim.x + threadIdx.x) >> 5;
  int lane  = threadIdx.x & 31;
  int base  = gwave * 16;
  if (base >= N) return;                 // wave-uniform exit: EXEC stays full
  int col   = lane & 15;                 // A: row M; B: col N
  int khalf = (lane >> 4) << 1;          // K sub-offset {0,2}
  int node  = base + col;
  // Clamp pointer only (no OOB read). Garbage rows land in store-guarded C rows.
  const float* xr = x + (size_t)(node < N ? node : 0) * 512;
  v8f c = {};
  for (int k = 0; k < 512; k += 4) {
    v2f a = *(const v2f*)(xr + k + khalf);          // 8B aligned (even index)
    v2f b;
    b.x = sW[(k + khalf) * 16 + col];               // ds_load
    b.y = sW[(k + khalf + 1) * 16 + col];
    c = __builtin_amdgcn_wmma_f32_16x16x4_f32(false, a, false, b, (short)0, c, false, false);
  }
  int mhi = (lane >> 4) << 3;            // C/D: M = mhi + i, Ncol = col
  for (int i = 0; i < 8; ++i) {
    int m = base + mhi + i;
    if (m < N) h1[(size_t)m * 16 + col] = c[i];
  }
}

// ---------------- GEMM2: h2[N,40] = o1[N,16] @ W2[16,40] via WMMA ----------------
// 3 N-tiles of 16 (cols 32..39 valid in last), 4 k-steps each; W2 in LDS.
__global__ void gemm2_wmma(const float* __restrict__ o1, const float* __restrict__ W2,
                           float* __restrict__ h2, int N) {
  __shared__ float sW[16 * 40];
  {
    unsigned lds_off = (unsigned)(uintptr_t)&sW[0];
    async_copy_to_lds(lds_off, W2, 16 * 40 * 4, threadIdx.x, blockDim.x);
  }
  __syncthreads();

  int gwave = (blockIdx.x * blockDim.x + threadIdx.x) >> 5;
  int lane  = threadIdx.x & 31;
  int base  = gwave * 16;
  if (base >= N) return;
  int lcol  = lane & 15;
  int khalf = (lane >> 4) << 1;
  int node  = base + lcol;
  const float* ar = o1 + (size_t)(node < N ? node : 0) * 16;
  v2f areg[4];
#pragma unroll
  for (int j = 0; j < 4; ++j) areg[j] = *(const v2f*)(ar + j * 4 + khalf);

  int mhi = (lane >> 4) << 3;
#pragma unroll
  for (int t = 0; t < 3; ++t) {
    int col = t * 16 + lcol;
    bool cval = col < 40;
    int bcol = cval ? col : 0;           // clamp LDS index; garbage cols unstored
    v8f c = {};
#pragma unroll
    for (int j = 0; j < 4; ++j) {
      v2f b;
      b.x = sW[(j * 4 + khalf) * 40 + bcol];
      b.y = sW[(j * 4 + khalf + 1) * 40 + bcol];
      c = __builtin_amdgcn_wmma_f32_16x16x4_f32(false, areg[j], false, b, (short)0, c, false, false);
    }
    if (cval) {
      for (int i = 0; i < 8; ++i) {
        int m = base + mhi + i;
        if (m < N) h2[(size_t)m * 40 + col] = c[i];
      }
    }
  }
}

// ---------------- per-node attention logits: a_s = h.att_src, a_d = h.att_dst ----------------
template<int F>
__global__ void attn_dots(const float* __restrict__ h, const float* __restrict__ att_s,
                          const float* __restrict__ att_d, float* __restrict__ a_s,
                          float* __restrict__ a_d, int N) {
  int n = blockIdx.x * blockDim.x + threadIdx.x;
  if (n >= N) return;
  const float* hr = h + (size_t)n * F;
  float s = 0.f, d = 0.f;
#pragma unroll
  for (int f = 0; f < F; ++f) { float v = hr[f]; s += v * att_s[f]; d += v * att_d[f]; }
  a_s[n] = s; a_d[n] = d;
}

// edge i < E: real edge (edge_index is int64 [2,E]); i >= E: self-loop (i-E).
__device__ __forceinline__ void edge_nodes(const long long* __restrict__ ei, int i, int E,
                                           int* s, int* d) {
  if (i < E) { *s = (int)ei[i]; *d = (int)ei[E + i]; }
  else       { *s = *d = i - E; }
}

__device__ __forceinline__ float edge_logit(const float* a_s, const float* a_d, int s, int d) {
  float v = a_s[s] + a_d[d];
  return v > 0.f ? v : NEG_SLOPE * v;   // leaky_relu
}

// ---------------- segment max over dst ----------------
__global__ void edge_max_k(const long long* __restrict__ ei, const float* __restrict__ a_s,
                           const float* __restrict__ a_d, float* __restrict__ m, int E, int N) {
  int i = blockIdx.x * blockDim.x + threadIdx.x;
  if (i >= E + N) return;
  int s, d; edge_nodes(ei, i, E, &s, &d);
  atomic_max_f32(&m[d], edge_logit(a_s, a_d, s, d));
}

// ---------------- segment sum of exp(e - m[dst]) ----------------
__global__ void edge_expsum_k(const long long* __restrict__ ei, const float* __restrict__ a_s,
                              const float* __restrict__ a_d, const float* __restrict__ m,
                              float* __restrict__ dn, int E, int N) {
  int i = blockIdx.x * blockDim.x + threadIdx.x;
  if (i >= E + N) return;
  int s, d; edge_nodes(ei, i, E, &s, &d);
  float v = edge_logit(a_s, a_d, s, d);
  atomicAdd(&dn[d], __expf(v - m[d]));
}

// ---------------- weighted aggregation: agg[dst] += alpha * h[src] ----------------
// One thread per (edge, feature): contiguous lanes share an edge -> broadcast
// index loads, coalesced h[src] gathers and agg[dst] atomics.
template<int F>
__global__ void edge_agg_k(const long long* __restrict__ ei, const float* __restrict__ a_s,
                           const float* __restrict__ a_d, const float* __restrict__ m,
                           const float* __restrict__ dn, const float* __restrict__ h,
                           float* __restrict__ agg, int E, int N) {
  long long idx = (long long)blockIdx.x * blockDim.x + threadIdx.x;
  long long total = (long long)(E + N) * F;
  if (idx >= total) return;
  int e = (int)(idx / F);
  int f = (int)(idx % F);
  int s, d; edge_nodes(ei, e, E, &s, &d);
  float v = edge_logit(a_s, a_d, s, d);
  float alpha = __expf(v - m[d]) / dn[d];
  atomicAdd(&agg[(size_t)d * F + f], alpha * h[(size_t)s * F + f]);
}

// ---------------- bias (+optional relu), in place ----------------
template<int F, bool RELU>
__global__ void finalize_k(float* __restrict__ a, const float* __restrict__ bias, int N) {
  int i = blockIdx.x * blockDim.x + threadIdx.x;
  if (i >= N * F) return;
  float v = a[i] + bias[i % F];
  if (RELU) v = v > 0.f ? v : 0.f;
  a[i] = v;
}

extern "C" void kernel_launch(void* const* d_in, const int* in_sizes, int n_in,
                              void* d_out, int out_size, void* d_ws, size_t ws_size,
                              hipStream_t stream) {
  const float* x   = (const float*)d_in[0];
  const float* W1  = (const float*)d_in[1];
  const float* as1 = (const float*)d_in[2];
  const float* ad1 = (const float*)d_in[3];
  const float* b1  = (const float*)d_in[4];
  const float* W2  = (const float*)d_in[5];
  const float* as2 = (const float*)d_in[6];
  const float* ad2 = (const float*)d_in[7];
  const float* b2  = (const float*)d_in[8];
  const long long* ei = (const long long*)d_in[9];   // int64 [2,E], row-major

  const int FIN = 512, H = 16, C = 40;
  int N = in_sizes[0] / FIN;
  int E = in_sizes[9] / 2;

  // workspace: 80*N floats = 32 MB
  float* p   = (float*)d_ws;
  float* h1  = p; p += (size_t)N * H;
  float* aS1 = p; p += N;
  float* aD1 = p; p += N;
  float* m1  = p; p += N;
  float* dn1 = p; p += N;
  float* o1  = p; p += (size_t)N * H;   // agg1, finalized in place to relu output
  float* h2  = p; p += (size_t)N * C;
  float* aS2 = p; p += N;
  float* aD2 = p; p += N;
  float* m2  = p; p += N;
  float* dn2 = p; p += N;
  float* out = (float*)d_out;

  const int B = 256;  // 8 waves per block
  auto cdiv = [](long long a, long long b) { return (int)((a + b - 1) / b); };

  gat_init<<<cdiv((long long)N * C, B), B, 0, stream>>>(m1, m2, dn1, dn2, o1, out, N);

  int ntiles = (N + 15) / 16;            // one wave per 16-node tile
  int gblocks = cdiv(ntiles, B / 32);

  // ---- layer 1 ----
  gemm1_wmma<<<gblocks, B, 0, stream>>>(x, W1, h1, N);
  attn_dots<16><<<cdiv(N, B), B, 0, stream>>>(h1, as1, ad1, aS1, aD1, N);
  edge_max_k   <<<cdiv(E + N, B), B, 0, stream>>>(ei, aS1, aD1, m1, E, N);
  edge_expsum_k<<<cdiv(E + N, B), B, 0, stream>>>(ei, aS1, aD1, m1, dn1, E, N);
  edge_agg_k<16><<<cdiv((long long)(E + N) * 16, B), B, 0, stream>>>(ei, aS1, aD1, m1, dn1, h1, o1, E, N);
  finalize_k<16, true><<<cdiv((long long)N * 16, B), B, 0, stream>>>(o1, b1, N);

  // ---- layer 2 ----
  gemm2_wmma<<<gblocks, B, 0, stream>>>(o1, W2, h2, N);
  attn_dots<40><<<cdiv(N, B), B, 0, stream>>>(h2, as2, ad2, aS2, aD2, N);
  edge_max_k   <<<cdiv(E + N, B), B, 0, stream>>>(ei, aS2, aD2, m2, E, N);
  edge_expsum_k<<<cdiv(E + N, B), B, 0, stream>>>(ei, aS2, aD2, m2, dn2, E, N);
  edge_agg_k<40><<<cdiv((long long)(E + N) * 40, B), B, 0, stream>>>(ei, aS2, aD2, m2, dn2, h2, out, E, N);
  finalize_k<40, false><<<cdiv((long long)N * 40, B), B, 0, stream>>>(out, b2, N);
}